// ImplicitQuadrupleExtractor_46935402610851
// MI455X (gfx1250) — compile-verified
//
#include <hip/hip_runtime.h>
#include <hip/hip_bf16.h>

// ---------------------------------------------------------------------------
// ImplicitQuadrupleExtractor for MI455X (gfx1250, wave32).
//   B=8, S=512, R=8.
//   Kernel 1: consistency MLP 8->32->16->1 via V_WMMA_F32_16X16X4_F32 tiles
//             (one wave per 16 tokens; 12 WMMAs/wave) + softmax flags.
//   Kernel 2: rule-4 triangular any-reduction over rel — the HBM-bound kernel.
//             Non-temporal b128 streaming + block-uniform early exit
//             (expected traffic ~14.6 MB instead of 34 MB for random data).
//   Kernel 3: per-token rule application + sequential isolation scan (b128 IO).
// ---------------------------------------------------------------------------

typedef __attribute__((ext_vector_type(2))) float v2f;
typedef __attribute__((ext_vector_type(4))) float v4f;
typedef __attribute__((ext_vector_type(8))) float v8f;

#define BB 8
#define SS 512
#define RR 8

__device__ __forceinline__ v8f wmma_f32_k4(v2f a, v2f b, v8f c) {
  // D = A(16x4 f32) * B(4x16 f32) + C(16x16 f32)
  return __builtin_amdgcn_wmma_f32_16x16x4_f32(
      /*neg_a=*/false, a, /*neg_b=*/false, b,
      /*c_mod=*/(short)0, c, /*reuse_a=*/false, /*reuse_b=*/false);
}

// ---------------------------------------------------------------------------
// Kernel 1: MLP score -> factor1, plus softmax implicit-prob flags pa/po.
// grid = (B*S)/16 blocks of 32 threads (1 wave = 1 tile of 16 tokens).
// ---------------------------------------------------------------------------
__global__ void __launch_bounds__(32)
mlp_score_kernel(const float* __restrict__ alog, const float* __restrict__ olog,
                 const float* __restrict__ W1, const float* __restrict__ b1,
                 const float* __restrict__ W2, const float* __restrict__ b2,
                 const float* __restrict__ W3, const float* __restrict__ b3,
                 float* __restrict__ factor1, int* __restrict__ paf,
                 int* __restrict__ pof) {
  const int tile = blockIdx.x;        // 256 tiles of 16 tokens
  const int lane = threadIdx.x;       // 0..31
  const int lr   = lane & 15;
  const int half = lane >> 4;
  const int t0   = tile * 16;

  __shared__ float h1[16][33];        // 16 tokens x 32 features (+pad)
  __shared__ float h2[16][17];        // 16 tokens x 16 features (+pad)

  // ---- A operands: X = concat(aspect, opinion)[16 tokens x 8 features] ----
  // 16x4 f32 A layout: lanes 0-15 rows M, VGPR0/1 = K=(2*half), K=(2*half+1)
  const int tA = t0 + lr;
  v2f a1, a2;
  a1.x = alog[tA * 4 + 2 * half + 0];
  a1.y = alog[tA * 4 + 2 * half + 1];
  a2.x = olog[tA * 4 + 2 * half + 0];      // features 4..7
  a2.y = olog[tA * 4 + 2 * half + 1];

  // ---- Layer 1: [16x8] @ W1[8x32] -> two 16x16 N-tiles ----
  // B 4x16 f32 layout: VGPR0 = row K=(2*half), VGPR1 = row K=(2*half+1), N=lr
  v2f bl0, bl1, bh0, bh1;
  bl0.x = W1[(2 * half + 0) * 32 + lr];
  bl0.y = W1[(2 * half + 1) * 32 + lr];
  bl1.x = W1[(4 + 2 * half + 0) * 32 + lr];
  bl1.y = W1[(4 + 2 * half + 1) * 32 + lr];
  bh0.x = W1[(2 * half + 0) * 32 + 16 + lr];
  bh0.y = W1[(2 * half + 1) * 32 + 16 + lr];
  bh1.x = W1[(4 + 2 * half + 0) * 32 + 16 + lr];
  bh1.y = W1[(4 + 2 * half + 1) * 32 + 16 + lr];

  v8f c_lo = {};
  v8f c_hi = {};
  c_lo = wmma_f32_k4(a1, bl0, c_lo);
  c_lo = wmma_f32_k4(a2, bl1, c_lo);
  c_hi = wmma_f32_k4(a1, bh0, c_hi);
  c_hi = wmma_f32_k4(a2, bh1, c_hi);

  // C layout: VGPR r, lane l -> row M = r + 8*half, col N = lr
#pragma unroll
  for (int r = 0; r < 8; ++r) {
    const int M = r + 8 * half;
    h1[M][lr]      = fmaxf(c_lo[r] + b1[lr], 0.0f);
    h1[M][16 + lr] = fmaxf(c_hi[r] + b1[16 + lr], 0.0f);
  }
  __syncthreads();

  // ---- Layer 2: H1[16x32] @ W2[32x16], 8 K-chunks of 4 ----
  v8f c2 = {};
#pragma unroll
  for (int kk = 0; kk < 8; ++kk) {
    v2f aa, bb;
    aa.x = h1[lr][kk * 4 + 2 * half + 0];
    aa.y = h1[lr][kk * 4 + 2 * half + 1];
    bb.x = W2[(kk * 4 + 2 * half + 0) * 16 + lr];
    bb.y = W2[(kk * 4 + 2 * half + 1) * 16 + lr];
    c2 = wmma_f32_k4(aa, bb, c2);
  }
#pragma unroll
  for (int r = 0; r < 8; ++r) {
    h2[r + 8 * half][lr] = fmaxf(c2[r] + b2[lr], 0.0f);
  }
  __syncthreads();

  // ---- Layer 3 + sigmoid + factor1 + softmax flags (lanes 0..15) ----
  if (half == 0) {
    const int t = t0 + lr;
    float s3 = b3[0];
#pragma unroll
    for (int f = 0; f < 16; ++f) s3 += h2[lr][f] * W3[f];
    const float score = 1.0f / (1.0f + __expf(-s3));
    factor1[t] = (score < 0.5f) ? 2.0f * score : 1.0f;

    // pa: softmax(aspect)[:2].sum > 0.5  <=>  e0+e1 > e2+e3
    {
      const v4f x = ((const v4f*)alog)[t];
      const float m  = fmaxf(fmaxf(x.x, x.y), fmaxf(x.z, x.w));
      const float e0 = __expf(x.x - m), e1 = __expf(x.y - m);
      const float e2 = __expf(x.z - m), e3 = __expf(x.w - m);
      paf[t] = (e0 + e1) > (e2 + e3) ? 1 : 0;
    }
    {
      const v4f x = ((const v4f*)olog)[t];
      const float m  = fmaxf(fmaxf(x.x, x.y), fmaxf(x.z, x.w));
      const float e0 = __expf(x.x - m), e1 = __expf(x.y - m);
      const float e2 = __expf(x.z - m), e3 = __expf(x.w - m);
      pof[t] = (e0 + e1) > (e2 + e3) ? 1 : 0;
    }
  }
}

// ---------------------------------------------------------------------------
// Kernel 2: rule 4 = any_{j<i, r}(rel[b,i,j,r] > 0.5).
// One block per (b,i) row. Chunked so every chunk is exactly one NT b128
// load per thread, followed by a barrier and a block-UNIFORM early-exit
// check (all threads read the same shared flag after the barrier, so the
// break cannot diverge). Max 4 chunks per row (n4 <= 1022).
// ---------------------------------------------------------------------------
__global__ void __launch_bounds__(256)
rule4_kernel(const float* __restrict__ rel, int* __restrict__ r4f) {
  const int row = blockIdx.x;              // b*S + i
  const int i   = row & (SS - 1);
  __shared__ int flag;
  if (threadIdx.x == 0) flag = 0;
  __syncthreads();

  const v4f* rp = (const v4f*)rel + (size_t)row * (SS * RR / 4);
  const int n4 = i * (RR / 4);             // only j < i
  for (int base = 0; base < n4; base += 256) {
    const int k = base + (int)threadIdx.x;
    if (k < n4) {
      const v4f v = __builtin_nontemporal_load(&rp[k]);  // single-use stream
      if ((v.x > 0.5f) || (v.y > 0.5f) || (v.z > 0.5f) || (v.w > 0.5f))
        flag = 1;                          // benign race: all writers store 1
    }
    __syncthreads();
    if (flag) break;                       // uniform after barrier
  }
  if (threadIdx.x == 0) r4f[row] = flag;
}

// ---------------------------------------------------------------------------
// Kernel 3: apply rules 1-4 to build base rows, then sequential isolation
// scan along S. One block per batch; b128 loads/stores throughout.
// ---------------------------------------------------------------------------
__global__ void __launch_bounds__(256)
apply_scan_kernel(const float* __restrict__ alog, const float* __restrict__ olog,
                  const int* __restrict__ ea, const int* __restrict__ eo,
                  const float* __restrict__ factor1, const int* __restrict__ paf,
                  const int* __restrict__ pof, const int* __restrict__ r4f,
                  float* __restrict__ out) {
  const int b = blockIdx.x;
  __shared__ v4f abase[SS];
  __shared__ v4f obase[SS];
  __shared__ unsigned char act_a[SS];      // act() of ORIGINAL rows
  __shared__ unsigned char act_o[SS];

  const v4f* a4 = (const v4f*)alog;
  const v4f* o4 = (const v4f*)olog;

  for (int s = threadIdx.x; s < SS; s += blockDim.x) {
    const int t = b * SS + s;
    const v4f a = a4[t];
    const v4f o = o4[t];
    const float f1 = factor1[t];

    bool near_op = false, near_as = false;
    const int lo = (s - 3 > 0) ? s - 3 : 0;
    const int hi = (s + 3 < SS - 1) ? s + 3 : SS - 1;
    for (int j = lo; j <= hi; ++j) {
      near_op |= (eo[b * SS + j] > 0);
      near_as |= (ea[b * SS + j] > 0);
    }
    const bool r2 = paf[t] && !near_op;
    const bool r3 = pof[t] && !near_as;
    const bool r4 = r4f[t] != 0;
    const float fa2 = (r2 ? 0.3f : 1.0f) * (r4 ? 0.7f : 1.0f);
    const float fo2 = (r3 ? 0.3f : 1.0f);

    v4f ab, ob;
    ab.x = a.x * f1 * fa2; ab.y = a.y * f1 * fa2;
    ab.z = a.z * f1;       ab.w = a.w * f1;
    ob.x = o.x * f1 * fo2; ob.y = o.y * f1 * fo2;
    ob.z = o.z * f1;       ob.w = o.w * f1;
    abase[s] = ab;
    obase[s] = ob;

    act_a[s] = (fmaxf(fmaxf(a.x, a.y), fmaxf(a.z, a.w)) > 0.5f) ? 1 : 0;
    act_o[s] = (fmaxf(fmaxf(o.x, o.y), fmaxf(o.z, o.w)) > 0.5f) ? 1 : 0;
  }
  __syncthreads();

  if (threadIdx.x == 0) {
    // carries: act() of previous FINAL rows; initial NEG rows -> false
    bool ap2 = false, ap1 = false, op2 = false, op1 = false;
    v4f* oa = (v4f*)out + (size_t)b * SS;
    v4f* oo = (v4f*)out + (size_t)gridDim.x * SS + (size_t)b * SS;
    for (int s = 0; s < SS; ++s) {
      const bool an1 = (s + 1 < SS) ? (act_a[s + 1] != 0) : false;
      const bool an2 = (s + 2 < SS) ? (act_a[s + 2] != 0) : false;
      const bool on1 = (s + 1 < SS) ? (act_o[s + 1] != 0) : false;
      const bool on2 = (s + 2 < SS) ? (act_o[s + 2] != 0) : false;
      const bool activity = ap2 | ap1 | op2 | op1 | an1 | an2 | on1 | on2;
      const float f = activity ? 1.0f : 0.1f;

      v4f af = abase[s];
      v4f of = obase[s];
      af.x *= f; af.y *= f;
      of.x *= f; of.y *= f;
      oa[s] = af;
      oo[s] = of;

      const bool acta = fmaxf(fmaxf(af.x, af.y), fmaxf(af.z, af.w)) > 0.5f;
      const bool acto = fmaxf(fmaxf(of.x, of.y), fmaxf(of.z, of.w)) > 0.5f;
      ap2 = ap1; ap1 = acta;
      op2 = op1; op1 = acto;
    }
  }
}

// ---------------------------------------------------------------------------
// Launch
// ---------------------------------------------------------------------------
extern "C" void kernel_launch(void* const* d_in, const int* in_sizes, int n_in,
                              void* d_out, int out_size, void* d_ws, size_t ws_size,
                              hipStream_t stream) {
  const float* alog = (const float*)d_in[0];   // [8,512,4]
  const float* olog = (const float*)d_in[1];   // [8,512,4]
  const float* rel  = (const float*)d_in[2];   // [8,512,512,8]
  const int*   ea   = (const int*)d_in[3];     // [8,512]
  const int*   eo   = (const int*)d_in[4];     // [8,512]
  const float* W1   = (const float*)d_in[5];   // [8,32]
  const float* b1   = (const float*)d_in[6];   // [32]
  const float* W2   = (const float*)d_in[7];   // [32,16]
  const float* b2   = (const float*)d_in[8];   // [16]
  const float* W3   = (const float*)d_in[9];   // [16,1]
  const float* b3   = (const float*)d_in[10];  // [1]
  float* out = (float*)d_out;

  const int BS = BB * SS;                      // 4096 tokens
  float* factor1 = (float*)d_ws;
  int*   paf     = (int*)((char*)d_ws + (size_t)BS * sizeof(float));
  int*   pof     = paf + BS;
  int*   r4f     = pof + BS;

  mlp_score_kernel<<<BS / 16, 32, 0, stream>>>(alog, olog, W1, b1, W2, b2, W3, b3,
                                               factor1, paf, pof);
  rule4_kernel<<<BS, 256, 0, stream>>>(rel, r4f);
  apply_scan_kernel<<<BB, 256, 0, stream>>>(alog, olog, ea, eo, factor1, paf, pof,
                                            r4f, out);
}